// TorchWLKernel_7284264534450
// MI455X (gfx1250) — compile-verified
//
#include <hip/hip_runtime.h>
#include <hip/hip_bf16.h>

// ---------------- problem constants ----------------
#define G_   64
#define N_   4096
#define D_   16
#define T_   4
#define GN_  (G_ * N_)            // 262144
#define INST_ ((T_ + 1) * GN_)    // 1310720 label instances total
#define SIZE_ (N_ + T_ * GN_)     // 1052672 max distinct ids
#define TAB_LOG_ 22
#define TAB_ (1u << TAB_LOG_)     // 4M slots, 32MB, L2-resident
#define CHUNK_ 65536              // id-range per Gram chunk
#define NCHUNK_ ((SIZE_ + CHUNK_ - 1) / CHUNK_)  // 17

typedef __attribute__((ext_vector_type(2))) float v2f;
typedef __attribute__((ext_vector_type(8))) float v8f;

__device__ __forceinline__ unsigned long long mix64(unsigned long long x) {
    x += 0x9E3779B97F4A7C15ull;
    x = (x ^ (x >> 30)) * 0xBF58476D1CE4E5B9ull;
    x = (x ^ (x >> 27)) * 0x94D049BB133111EBull;
    return x ^ (x >> 31);
}

// -------- init: clear hash table, counter, K accumulator, labels0 = arange --------
__global__ void wl_init_kernel(unsigned long long* __restrict__ keys,
                               int* __restrict__ counter,
                               float* __restrict__ kacc,
                               int* __restrict__ lab0) {
    unsigned i = blockIdx.x * blockDim.x + threadIdx.x;
    if (i < TAB_) keys[i] = 0ull;
    if (i == 0) *counter = N_;
    if (i < G_ * G_) kacc[i] = 0.0f;
    if (i < GN_) lab0[i] = (int)(i & (N_ - 1));
}

// -------- one WL relabeling round: gather, multiset-hash, dedup via global hash table --------
__global__ void wl_step_kernel(const int* __restrict__ nbr,
                               const int* __restrict__ prev,
                               int* __restrict__ cur,
                               unsigned long long* __restrict__ keys,
                               int* __restrict__ counter,
                               unsigned salt) {
    int tid = blockIdx.x * blockDim.x + threadIdx.x;   // tid = g*N + n
    int g = tid >> 12;
    const int* pl = prev + (g << 12);

    // 64B contiguous neighbor-index load per lane (4 x int4)
    const int4* nb4 = (const int4*)(nbr + ((size_t)tid << 4));
    int idx[D_];
    int4 q0 = nb4[0], q1 = nb4[1], q2 = nb4[2], q3 = nb4[3];
    idx[0]=q0.x; idx[1]=q0.y; idx[2]=q0.z; idx[3]=q0.w;
    idx[4]=q1.x; idx[5]=q1.y; idx[6]=q1.z; idx[7]=q1.w;
    idx[8]=q2.x; idx[9]=q2.y; idx[10]=q2.z; idx[11]=q2.w;
    idx[12]=q3.x; idx[13]=q3.y; idx[14]=q3.z; idx[15]=q3.w;

    int own = pl[tid & (N_ - 1)];
    unsigned long long hsum = 0ull;
    bool keep = true;
#pragma unroll
    for (int d = 0; d < D_; ++d) {
        int lab = pl[idx[d]];                 // L2-hot gather (16KB window)
        keep = keep && (lab == own);
        hsum += mix64((unsigned long long)(unsigned)lab);  // commutative multiset hash
    }

    int newlab;
    if (keep) {
        newlab = own;                         // reference's "keep old label" rule
    } else {
        unsigned long long h =
            mix64(hsum ^ mix64(((unsigned long long)salt << 32) |
                               (unsigned long long)(unsigned)own));
        unsigned key32 = (unsigned)(h >> 32) | 0x80000000u;  // never-zero entry
        unsigned slot = (unsigned)h & (TAB_ - 1);
        int myid = -1;
        for (;;) {
            unsigned long long e = *(volatile unsigned long long*)&keys[slot];
            if (e == 0ull) {
                if (myid < 0) myid = atomicAdd(counter, 1);  // pre-alloc id (waste-safe)
                unsigned long long want =
                    ((unsigned long long)key32 << 32) | (unsigned)myid;
                unsigned long long old = atomicCAS(&keys[slot], 0ull, want);
                if (old == 0ull) { newlab = myid; break; }   // key+id published atomically
                e = old;
            }
            if ((unsigned)(e >> 32) == key32) { newlab = (int)(unsigned)e; break; }
            slot = (slot + 1) & (TAB_ - 1);
        }
    }
    cur[tid] = newlab;
}

// -------- zero one F chunk tile: [CHUNK][64] f32, 16MB, stays in L2 --------
__global__ void wl_zero_kernel(float* __restrict__ fch) {
    unsigned i = blockIdx.x * blockDim.x + threadIdx.x;
    fch[i] = 0.0f;
}

// -------- scatter-add node weights for ids falling in [base, base+CHUNK) --------
__global__ void wl_scatter_kernel(const int* __restrict__ lall,
                                  const float* __restrict__ w,
                                  float* __restrict__ fch, int base) {
    int i = blockIdx.x * blockDim.x + threadIdx.x;   // over (T+1)*G*N instances
    int id = lall[i];
    unsigned rel = (unsigned)(id - base);
    if (rel < (unsigned)CHUNK_) {
        int r = i & (GN_ - 1);          // g*N+n
        int g = r >> 12;
        atomicAdd(&fch[(size_t)rel * G_ + g], w[r]);
    }
}

// -------- Gram accumulate: Kacc += Fchunkᵀ-view product via V_WMMA_F32_16X16X4_F32 --------
// fch layout: [k][64] (k-major). 32 blocks split the 64K k-range; 16 waves = 4x4 tiles of K.
__global__ void __launch_bounds__(512)
wl_gram_kernel(const float* __restrict__ fch, float* __restrict__ kacc) {
    int wave = threadIdx.x >> 5;          // wave32
    int lane = threadIdx.x & 31;
    int ti = wave >> 2, tj = wave & 3;    // 16x16 tile (ti,tj) of 64x64 K
    int m15 = lane & 15;
    int khalf = (lane < 16) ? 0 : 2;      // A/B f32 layout: VGPR0 holds K0|K2, VGPR1 K1|K3
    int rowA = ti * 16 + m15;
    int rowB = tj * 16 + m15;
    int kbase = blockIdx.x * (CHUNK_ / 32);   // 2048 k-steps per block

    v8f c = {0.f, 0.f, 0.f, 0.f, 0.f, 0.f, 0.f, 0.f};
    for (int k = kbase; k < kbase + (CHUNK_ / 32); k += 4) {
        int kk = k + khalf;
        v2f a, b;
        a.x = fch[(size_t)kk * G_ + rowA];
        a.y = fch[(size_t)(kk + 1) * G_ + rowA];
        b.x = fch[(size_t)kk * G_ + rowB];
        b.y = fch[(size_t)(kk + 1) * G_ + rowB];
        c = __builtin_amdgcn_wmma_f32_16x16x4_f32(
                false, a, false, b, (short)0, c, false, false);
    }
    // C/D layout: VGPR v -> M=v (lanes 0-15) / M=v+8 (lanes 16-31), N=lane&15
    int Nn = lane & 15;
    int moff = (lane < 16) ? 0 : 8;
#pragma unroll
    for (int v = 0; v < 8; ++v) {
        atomicAdd(&kacc[(ti * 16 + v + moff) * G_ + tj * 16 + Nn], c[v]);
    }
}

// -------- normalize: out[g,h] = K[g,h] / sqrt(K[g,g]*K[h,h]) --------
__global__ void wl_norm_kernel(const float* __restrict__ kacc,
                               float* __restrict__ out) {
    int i = blockIdx.x * blockDim.x + threadIdx.x;  // 4096
    int g = i >> 6, h = i & 63;
    float dg = kacc[g * G_ + g];
    float dh = kacc[h * G_ + h];
    out[i] = kacc[i] / sqrtf(dg * dh);
}

extern "C" void kernel_launch(void* const* d_in, const int* in_sizes, int n_in,
                              void* d_out, int out_size, void* d_ws, size_t ws_size,
                              hipStream_t stream) {
    const int* nbr = (const int*)d_in[0];           // [G,N,D] int32
    const float* w = (const float*)d_in[1];         // [G,N] float32
    float* out = (float*)d_out;                     // [G,G] float32

    // workspace carve-out (~56 MB)
    unsigned char* p = (unsigned char*)d_ws;
    int* lall = (int*)p;                            p += (size_t)INST_ * sizeof(int);
    unsigned long long* keys = (unsigned long long*)p; p += (size_t)TAB_ * 8;
    int* counter = (int*)p;                         p += 256;
    float* kacc = (float*)p;                        p += (size_t)G_ * G_ * sizeof(float);
    float* fch = (float*)p;                         // CHUNK_*64 floats = 16MB

    // init: table + counter + Kacc + labels[0]
    wl_init_kernel<<<TAB_ / 256, 256, 0, stream>>>(keys, counter, kacc, lall);

    // 4 WL rounds (global sync between rounds via kernel boundaries)
    for (int t = 1; t <= T_; ++t) {
        wl_step_kernel<<<GN_ / 256, 256, 0, stream>>>(
            nbr, lall + (size_t)(t - 1) * GN_, lall + (size_t)t * GN_,
            keys, counter, (unsigned)t * 0x85EBCA6Bu);
    }

    // K = F F^T streamed over id chunks, WMMA-accumulated
    for (int c = 0; c < NCHUNK_; ++c) {
        wl_zero_kernel<<<(CHUNK_ * G_) / 256, 256, 0, stream>>>(fch);
        wl_scatter_kernel<<<INST_ / 256, 256, 0, stream>>>(lall, w, fch, c * CHUNK_);
        wl_gram_kernel<<<32, 512, 0, stream>>>(fch, kacc);
    }

    wl_norm_kernel<<<(G_ * G_) / 256, 256, 0, stream>>>(kacc, out);
}